// Encoder_61572651155698
// MI455X (gfx1250) — compile-verified
//
#include <hip/hip_runtime.h>
#include <cstdint>
#include <cstddef>

#define NN    100000   // nodes
#define NE    800000   // edges
#define KIN   128      // layer0 input width
#define KHID  64       // hidden / layer1 input width
#define HEADS 3
#define OUTC  (HEADS*KHID)   // 192
#define NSLOPE 0.2f

typedef float v2f __attribute__((ext_vector_type(2)));
typedef float v8f __attribute__((ext_vector_type(8)));

// ---------- helpers ----------
__device__ __forceinline__ unsigned fkey(float f) {
    unsigned u = __float_as_uint(f);
    return (u & 0x80000000u) ? ~u : (u | 0x80000000u);
}
__device__ __forceinline__ float unfkey(unsigned k) {
    return (k & 0x80000000u) ? __uint_as_float(k ^ 0x80000000u)
                             : __uint_as_float(~k);
}

// ---------- zero fill ----------
__global__ void k_zero(float* __restrict__ p, size_t n) {
    size_t i = (size_t)blockIdx.x * blockDim.x + threadIdx.x;
    size_t stride = (size_t)gridDim.x * blockDim.x;
    for (; i < n; i += stride) p[i] = 0.f;
}

// ---------- Z[N,192] = X[N,K] @ W[K,192] via V_WMMA_F32_16X16X4_F32 ----------
// One wave per 16-row M-tile; A-fragment reused across the 12 N-tiles.
__global__ void k_gemm(const float* __restrict__ X, const float* __restrict__ W,
                       float* __restrict__ Z, int n, int K) {
    int wave = (int)(((size_t)blockIdx.x * blockDim.x + threadIdx.x) >> 5);
    int lane = threadIdx.x & 31;
    int mtiles = n >> 4;
    if (wave >= mtiles) return;          // wave-uniform: EXEC stays all-ones

    const int m0   = wave << 4;
    const int half = lane >> 4;          // 0 => K=k0,k0+1 ; 1 => K=k0+2,k0+3
    const int l16  = lane & 15;

    v8f acc[12];
#pragma unroll
    for (int t = 0; t < 12; ++t) { v8f z0 = {}; acc[t] = z0; }

    const float* xrow = X + (size_t)(m0 + l16) * K;
    for (int k0 = 0; k0 < K; k0 += 4) {
        const int ka = k0 + half * 2;
        v2f a;
        a.x = xrow[ka];
        a.y = xrow[ka + 1];
        const float* wrow = W + (size_t)ka * OUTC;
#pragma unroll
        for (int t = 0; t < 12; ++t) {
            v2f b;
            b.x = wrow[t * 16 + l16];          // K = ka,   N = n0+l16
            b.y = wrow[OUTC + t * 16 + l16];   // K = ka+1, N = n0+l16
            acc[t] = __builtin_amdgcn_wmma_f32_16x16x4_f32(
                false, a, false, b, (short)0, acc[t], false, false);
        }
    }
    // C/D layout: VGPR v -> row m0+v (lanes 0-15) / m0+8+v (lanes 16-31), col n0+l16
#pragma unroll
    for (int t = 0; t < 12; ++t) {
        const int n0 = t * 16;
#pragma unroll
        for (int v = 0; v < 8; ++v) {
            const int row = m0 + v + half * 8;
            Z[(size_t)row * OUTC + n0 + l16] = acc[t][v];
        }
    }
}

// ---------- el/er per (node, head) ----------
__global__ void k_attn_lr(const float* __restrict__ Z, const float* __restrict__ al,
                          const float* __restrict__ ar, float* __restrict__ el,
                          float* __restrict__ er) {
    int idx = blockIdx.x * blockDim.x + threadIdx.x;
    if (idx >= NN * HEADS) return;
    int node = idx / HEADS, h = idx - node * HEADS;
    const float* zp  = Z + (size_t)node * OUTC + h * KHID;
    const float* alp = al + h * KHID;
    const float* arp = ar + h * KHID;
    float sl = 0.f, sr = 0.f;
    for (int f = 0; f < KHID; ++f) {
        float zv = zp[f];
        sl += zv * alp[f];
        sr += zv * arp[f];
    }
    el[idx] = sl;
    er[idx] = sr;
}

// ---------- pass 1: e = leaky(el[src]+er[dst]); segment max over dst ----------
__global__ void k_edge_max(const int* __restrict__ src, const int* __restrict__ dst,
                           const float* __restrict__ el, const float* __restrict__ er,
                           float* __restrict__ ea, unsigned* __restrict__ mkey) {
    int idx = blockIdx.x * blockDim.x + threadIdx.x;
    if (idx >= NE * HEADS) return;
    int e = idx / HEADS, h = idx - e * HEADS;
    int s = src[e], d = dst[e];
    float v = el[s * HEADS + h] + er[d * HEADS + h];
    v = (v > 0.f) ? v : NSLOPE * v;
    ea[idx] = v;
    atomicMax(&mkey[d * HEADS + h], fkey(v));
}

// ---------- pass 2: a = exp(e - m[dst]); segment sum ----------
__global__ void k_edge_exp(const int* __restrict__ dst, float* __restrict__ ea,
                           const unsigned* __restrict__ mkey, float* __restrict__ ssum) {
    int idx = blockIdx.x * blockDim.x + threadIdx.x;
    if (idx >= NE * HEADS) return;
    int e = idx / HEADS, h = idx - e * HEADS;
    int d = dst[e];
    float m = unfkey(mkey[d * HEADS + h]);
    float a = expf(ea[idx] - m);
    ea[idx] = a;
    atomicAdd(&ssum[d * HEADS + h], a);
}

// ---------- pass 3: agg[dst] += alpha * z[src], one wave per edge ----------
__global__ void k_edge_msg(const int* __restrict__ src, const int* __restrict__ dst,
                           const float* __restrict__ ea, const float* __restrict__ ssum,
                           const float* __restrict__ Z, float* __restrict__ agg) {
    int wave = (int)(((size_t)blockIdx.x * blockDim.x + threadIdx.x) >> 5);
    int lane = threadIdx.x & 31;
    if (wave >= NE) return;
    int s = src[wave], d = dst[wave];
    float alpha[HEADS];
#pragma unroll
    for (int h = 0; h < HEADS; ++h)
        alpha[h] = ea[wave * HEADS + h] / ssum[d * HEADS + h];
    const float* zs = Z + (size_t)s * OUTC;
    float* ad = agg + (size_t)d * OUTC;
#pragma unroll
    for (int j = 0; j < 6; ++j) {            // 6*32 = 192 elems; head = j>>1
        int f = lane + 32 * j;
        atomicAdd(&ad[f], alpha[j >> 1] * zs[f]);
    }
}

// ---------- out[n,64] = mean over heads of (agg + bias) ----------
__global__ void k_headmean(const float* __restrict__ agg, const float* __restrict__ b,
                           float* __restrict__ out) {
    int idx = blockIdx.x * blockDim.x + threadIdx.x;
    if (idx >= NN * KHID) return;
    int node = idx >> 6, f = idx & 63;
    const float* ap = agg + (size_t)node * OUTC;
    float v = (ap[f] + b[f]) + (ap[KHID + f] + b[KHID + f]) +
              (ap[2 * KHID + f] + b[2 * KHID + f]);
    out[idx] = v * (1.f / 3.f);
}

// ---------- column-sum reduction over nodes (two-stage) ----------
__global__ void k_colsum(const float* __restrict__ H, float* __restrict__ colsum) {
    int f = threadIdx.x;                      // blockDim.x == 64
    float s = 0.f;
    for (int n = blockIdx.x; n < NN; n += gridDim.x)
        s += H[(size_t)n * KHID + f];
    atomicAdd(&colsum[f], s);
}

__global__ void k_writeout(const float* __restrict__ colsum, float* __restrict__ out) {
    int f = threadIdx.x;
    if (f < KHID) out[f] = colsum[f] * (1.f / (float)NN);
}

// ---------- one GAT layer ----------
static void run_layer(const float* X, int K, const float* W, const float* al,
                      const float* ar, const float* b,
                      const int* src, const int* dst,
                      float* Z, float* agg, float* el, float* er,
                      unsigned* mkey, float* ssum, float* ea,
                      float* out, hipStream_t stream) {
    const int B = 256;
    const int mtiles = NN / 16;
    const int gblocks = (mtiles * 32 + B - 1) / B;

    k_gemm<<<gblocks, B, 0, stream>>>(X, W, Z, NN, K);
    k_attn_lr<<<(NN * HEADS + B - 1) / B, B, 0, stream>>>(Z, al, ar, el, er);

    k_zero<<<512, B, 0, stream>>>((float*)mkey, (size_t)NN * HEADS);
    k_zero<<<512, B, 0, stream>>>(ssum, (size_t)NN * HEADS);
    k_zero<<<4096, B, 0, stream>>>(agg, (size_t)NN * OUTC);

    k_edge_max<<<(NE * HEADS + B - 1) / B, B, 0, stream>>>(src, dst, el, er, ea, mkey);
    k_edge_exp<<<(NE * HEADS + B - 1) / B, B, 0, stream>>>(dst, ea, mkey, ssum);
    k_edge_msg<<<(int)(((size_t)NE * 32 + B - 1) / B), B, 0, stream>>>(src, dst, ea, ssum, Z, agg);

    k_headmean<<<(NN * KHID + B - 1) / B, B, 0, stream>>>(agg, b, out);
}

extern "C" void kernel_launch(void* const* d_in, const int* in_sizes, int n_in,
                              void* d_out, int out_size, void* d_ws, size_t ws_size,
                              hipStream_t stream) {
    const float* h   = (const float*)d_in[0];
    const int*   src = (const int*)  d_in[1];
    const int*   dst = (const int*)  d_in[2];
    const float* W0  = (const float*)d_in[3];
    const float* al0 = (const float*)d_in[4];
    const float* ar0 = (const float*)d_in[5];
    const float* b0  = (const float*)d_in[6];
    const float* W1  = (const float*)d_in[7];
    const float* al1 = (const float*)d_in[8];
    const float* ar1 = (const float*)d_in[9];
    const float* b1  = (const float*)d_in[10];

    // workspace carve-up (floats)
    float* ws = (float*)d_ws;
    float* Z      = ws;               ws += (size_t)NN * OUTC;   // 19.2M
    float* agg    = ws;               ws += (size_t)NN * OUTC;   // 19.2M
    float* h1     = ws;               ws += (size_t)NN * KHID;   // 6.4M
    float* el     = ws;               ws += (size_t)NN * HEADS;
    float* er     = ws;               ws += (size_t)NN * HEADS;
    unsigned* mkey = (unsigned*)ws;   ws += (size_t)NN * HEADS;
    float* ssum   = ws;               ws += (size_t)NN * HEADS;
    float* ea     = ws;               ws += (size_t)NE * HEADS;  // 2.4M
    float* colsum = ws;               ws += 64;
    float* h2     = Z;   // safe reuse: k_headmean reads only agg/bias

    // layer 0: 128 -> 64 (3 heads, head-mean)
    run_layer(h, KIN, W0, al0, ar0, b0, src, dst,
              Z, agg, el, er, mkey, ssum, ea, h1, stream);
    // layer 1: 64 -> 64
    run_layer(h1, KHID, W1, al1, ar1, b1, src, dst,
              Z, agg, el, er, mkey, ssum, ea, h2, stream);

    // graph readout: mean over all nodes -> [64]
    k_zero<<<1, 64, 0, stream>>>(colsum, 64);
    k_colsum<<<512, 64, 0, stream>>>(h2, colsum);
    k_writeout<<<1, 64, 0, stream>>>(colsum, (float*)d_out);
}